// MambaBlock_43739946943084
// MI455X (gfx1250) — compile-verified
//
#include <hip/hip_runtime.h>
#include <hip/hip_bf16.h>

// MambaBlock on gfx1250 (MI455X). All GEMMs via V_WMMA_F32_16X16X4_F32
// (fp32-exact matrix path, wave32). Each wave computes a 16x64 output strip
// (4 tiles) reusing the A fragment 4x per K-step -> 1.25 b64 loads per WMMA
// instead of 2, and 4 independent accumulators for XDL pipelining.
// Einsum over S collapsed to per-row dot products (see analysis).

#define BB   2
#define MM   4
#define LL   256
#define DD   256
#define DD2  512
#define SS   16
#define RTOT 2048          // B*M*L
#define MLTOT 1024         // M*L

typedef float v2f __attribute__((ext_vector_type(2)));
typedef float v8f __attribute__((ext_vector_type(8)));

__device__ __forceinline__ float silu_f(float x) {
    return x / (1.0f + __expf(-x));
}
__device__ __forceinline__ float softplus_f(float x) {
    return (x > 20.0f) ? x : log1pf(__expf(x));
}

// ---------------------------------------------------------------------------
// RMSNorm: one wave per row of D=256.
// ---------------------------------------------------------------------------
__global__ void rmsnorm_kernel(const float* __restrict__ x,
                               const float* __restrict__ w,
                               float* __restrict__ xn) {
    int wave = (int)((blockIdx.x * blockDim.x + threadIdx.x) >> 5);
    int lane = (int)(threadIdx.x & 31);
    if (wave >= RTOT) return;
    const float* row = x + (size_t)wave * DD;
    float ss = 0.0f;
#pragma unroll
    for (int i = 0; i < DD / 32; ++i) {
        float v = row[lane + i * 32];
        ss += v * v;
    }
#pragma unroll
    for (int off = 16; off > 0; off >>= 1) ss += __shfl_xor(ss, off, 32);
    float scale = rsqrtf(ss * (1.0f / DD) + 1e-5f);
#pragma unroll
    for (int i = 0; i < DD / 32; ++i) {
        int c = lane + i * 32;
        xn[(size_t)wave * DD + c] = row[c] * scale * w[c];
    }
}

// ---------------------------------------------------------------------------
// WMMA fp32 GEMM: out[r][n] = act( sum_k A[r][k]*W[n][k] + bias[n] )
// A: RTOT x K row-major.  W: N x K row-major (torch (out,in) layout).
// Wave computes a 16 x (16*NT) strip; K-loop in steps of 4 via
// v_wmma_f32_16x16x4_f32. Fragment layouts per CDNA5 ISA 7.12.2:
//   A 16x4 : lanes 0-15 rows 0-15 {K=0,1}; lanes 16-31 rows 0-15 {K=2,3}
//   B 4x16 : lanes 0-15 cols 0-15 {K=0,1}; lanes 16-31 cols 0-15 {K=2,3}
//   C/D    : vgpr g, lane ln -> row = 8*(ln>>4)+g, col = ln&15
// ---------------------------------------------------------------------------
template <int KDIM, int NDIM, int ACT>
__global__ void wmma_gemm_kernel(const float* __restrict__ A,
                                 const float* __restrict__ W,
                                 const float* __restrict__ bias,
                                 float* __restrict__ Cout) {
    constexpr int NT = (NDIM >= 64) ? 4 : 1;      // N-tiles per wave
    constexpr int tiles_ng = NDIM / (16 * NT);
    constexpr int total = (RTOT / 16) * tiles_ng;
    int waveId = (int)((blockIdx.x * blockDim.x + threadIdx.x) >> 5);
    if (waveId >= total) return;                  // wave-uniform guard
    int tm = waveId / tiles_ng;
    int tg = waveId % tiles_ng;
    int lane = (int)(threadIdx.x & 31);
    int half = lane >> 4;                // 0: K pair {0,1}, 1: K pair {2,3}
    int lr   = lane & 15;                // row (A) / col (B) within tile

    const float* arow  = A + (size_t)(tm * 16 + lr) * KDIM;
    const float* wbase = W + (size_t)(tg * NT * 16 + lr) * KDIM;

    v8f acc[NT];
#pragma unroll
    for (int j = 0; j < NT; ++j) acc[j] = (v8f){0.f,0.f,0.f,0.f,0.f,0.f,0.f,0.f};

    for (int k0 = 0; k0 < KDIM; k0 += 16) {
        if (k0 + 128 < KDIM) {           // pull next chunk toward L2/L0
            __builtin_prefetch(arow + k0 + 128, 0, 1);
#pragma unroll
            for (int j = 0; j < NT; ++j)
                __builtin_prefetch(wbase + (size_t)j * 16 * KDIM + k0 + 128, 0, 1);
        }
#pragma unroll
        for (int kk = 0; kk < 16; kk += 4) {
            int ka = k0 + kk + 2 * half;
            v2f a = *(const v2f*)(arow + ka);     // contiguous 8B load, reused NT times
#pragma unroll
            for (int j = 0; j < NT; ++j) {
                v2f b = *(const v2f*)(wbase + (size_t)j * 16 * KDIM + ka);
                acc[j] = __builtin_amdgcn_wmma_f32_16x16x4_f32(
                    false, a, false, b, (short)0, acc[j], false, false);
            }
        }
    }

    int orow0 = tm * 16 + half * 8;
#pragma unroll
    for (int j = 0; j < NT; ++j) {
        int col = tg * NT * 16 + j * 16 + lr;
        float bn = bias[col];
#pragma unroll
        for (int g = 0; g < 8; ++g) {
            float v = acc[j][g] + bn;
            if (ACT == 1) v = silu_f(v);
            Cout[(size_t)(orow0 + g) * NDIM + col] = v;
        }
    }
}

// ---------------------------------------------------------------------------
// vb = flip(mouth_p) along L and/or D2.  Layout (B,M,L,D2).
// ---------------------------------------------------------------------------
__global__ void flip_copy_kernel(const float* __restrict__ src,
                                 float* __restrict__ dst,
                                 int flipL, int flipD) {
    int idx = (int)(blockIdx.x * blockDim.x + threadIdx.x);  // 2^20 threads
    int d  = idx & (DD2 - 1);
    int t  = idx >> 9;
    int l  = t & (LL - 1);
    int bm = t >> 8;                    // b*M + m
    int sl = flipL ? (LL - 1 - l) : l;
    int sd = flipD ? (DD2 - 1 - d) : d;
    dst[idx] = src[((size_t)(bm * LL + sl) << 9) | sd];
}

// ---------------------------------------------------------------------------
// hc = silu( conv2d_3x3_SAME( flip(x_proj) ) + conv_b ), NCHW with C=M=4,
// H=L, W=D2. Flips folded into the gather.
// ---------------------------------------------------------------------------
__global__ void conv_silu_kernel(const float* __restrict__ xproj,
                                 const float* __restrict__ cw,
                                 const float* __restrict__ cbias,
                                 float* __restrict__ hc,
                                 int flipL, int flipD) {
    int idx = (int)(blockIdx.x * blockDim.x + threadIdx.x);  // 2^20 threads
    int d  = idx & 511;
    int l  = (idx >> 9) & 255;
    int mo = (idx >> 17) & 3;
    int b  = idx >> 19;
    float acc = cbias[mo];
#pragma unroll
    for (int mi = 0; mi < MM; ++mi) {
        const float* base = xproj + (((size_t)(b * MM + mi)) << 17); // *L*D2
        const float* wk = cw + ((mo * MM + mi) * 9);
#pragma unroll
        for (int kh = 0; kh < 3; ++kh) {
            int ll = l + kh - 1;
            if ((unsigned)ll >= LL) continue;
            int sl = flipL ? (LL - 1 - ll) : ll;
#pragma unroll
            for (int kw = 0; kw < 3; ++kw) {
                int dd = d + kw - 1;
                if ((unsigned)dd >= DD2) continue;
                int sd = flipD ? (DD2 - 1 - dd) : dd;
                acc += base[((size_t)sl << 9) | sd] * wk[kh * 3 + kw];
            }
        }
    }
    hc[idx] = silu_f(acc);
}

// ---------------------------------------------------------------------------
// cb[r] = <Cm[r], Bm[r]>, ce[r] = <Cm[r], E[r]>  over S=16.
// ---------------------------------------------------------------------------
__global__ void dots_kernel(const float* __restrict__ cm,
                            const float* __restrict__ bm,
                            const float* __restrict__ e,
                            float* __restrict__ cb,
                            float* __restrict__ ce) {
    int r = (int)(blockIdx.x * blockDim.x + threadIdx.x);
    if (r >= RTOT) return;
    float s1 = 0.f, s2 = 0.f;
#pragma unroll
    for (int s = 0; s < SS; ++s) {
        float c = cm[r * SS + s];
        s1 += c * bm[r * SS + s];
        s2 += c * e[r * SS + s];
    }
    cb[r] = s1;
    ce[r] = s2;
}

// ---------------------------------------------------------------------------
// Per-branch fused S6 epilogue, accumulated into `combined` ((B,M,L,D2) rows).
// Row r=(b,m,l) pairs with row q=(b,l,m) in the (B,L,M)-viewed linears:
//   y = softplus(g1x[q][d]) * (hc[r][d]*cb[q] + vb[r][d]*ce[q])
//   contribution = silu( y * (g1x[q][d] + g1v[q][d]) )
// ---------------------------------------------------------------------------
__global__ void combine_kernel(const float* __restrict__ g1x,
                               const float* __restrict__ g1v,
                               const float* __restrict__ hc,
                               const float* __restrict__ vb,
                               const float* __restrict__ cb,
                               const float* __restrict__ ce,
                               float* __restrict__ combined,
                               int first) {
    int idx = (int)(blockIdx.x * blockDim.x + threadIdx.x);  // 2^20
    int d = idx & 511;
    int r = idx >> 9;
    int b = r >> 10;           // / (M*L)
    int t = r & 1023;
    int m = t >> 8;            // / L
    int l = t & 255;
    int q = b * MLTOT + l * MM + m;
    float gx = g1x[((size_t)q << 9) | d];
    float gv = g1v[((size_t)q << 9) | d];
    float y = softplus_f(gx) * (hc[idx] * cb[q] + vb[idx] * ce[q]);
    float val = silu_f(y * (gx + gv));
    if (first) combined[idx] = val;
    else       combined[idx] += val;
}

// combined *= 0.25 * x_residual  (both (B,M,L,D2))
__global__ void scale_kernel(float* __restrict__ combined,
                             const float* __restrict__ xres) {
    int idx = (int)(blockIdx.x * blockDim.x + threadIdx.x);
    combined[idx] *= 0.25f * xres[idx];
}

// ---------------------------------------------------------------------------
extern "C" void kernel_launch(void* const* d_in, const int* in_sizes, int n_in,
                              void* d_out, int out_size, void* d_ws, size_t ws_size,
                              hipStream_t stream) {
    (void)in_sizes; (void)n_in; (void)out_size; (void)ws_size;
    const float* x      = (const float*)d_in[0];
    const float* mouth  = (const float*)d_in[1];
    const float* w_inp  = (const float*)d_in[2];
    const float* b_inp  = (const float*)d_in[3];
    const float* w_out  = (const float*)d_in[4];
    const float* b_out  = (const float*)d_in[5];
    const float* w_d    = (const float*)d_in[6];
    const float* b_d    = (const float*)d_in[7];
    const float* conv_w = (const float*)d_in[8];
    const float* conv_b = (const float*)d_in[9];
    const float* fc1_w  = (const float*)d_in[10];
    const float* fc1_b  = (const float*)d_in[11];
    const float* fc2_w  = (const float*)d_in[12];
    const float* fc2_b  = (const float*)d_in[13];
    const float* fc3_w  = (const float*)d_in[14];
    const float* fc3_b  = (const float*)d_in[15];
    // d_in[16] = A_mat: multiplies an all-zero h buffer in the reference -> unused.
    const float* norm_w = (const float*)d_in[17];
    float* out = (float*)d_out;

    // Workspace layout (floats), ~36 MB total
    float* ws = (float*)d_ws;
    float* xn       = ws;                      // 2048*256
    float* xproj    = xn + RTOT * DD;          // 2048*512
    float* mouthp   = xproj + RTOT * DD2;
    float* xres     = mouthp + RTOT * DD2;
    float* combined = xres + RTOT * DD2;
    float* hc       = combined + RTOT * DD2;
    float* vb       = hc + RTOT * DD2;
    float* g1x      = vb + RTOT * DD2;
    float* g1v      = g1x + RTOT * DD2;
    float* bmrows   = g1v + RTOT * DD2;        // 2048*16
    float* erows    = bmrows + RTOT * SS;
    float* cmrows   = erows + RTOT * SS;
    float* cbv      = cmrows + RTOT * SS;      // 2048
    float* cev      = cbv + RTOT;

    const int TB = 256;
    const int ELEM_BLOCKS = (RTOT * DD2) / TB;                       // 4096
    // waves = (RTOT/16) * (N / (16*NT)); NT=4 for N>=64, NT=1 for N=16
    const int GEMM512_BLOCKS = ((RTOT / 16) * (DD2 / 64) * 32) / TB; // 128
    const int GEMM256_BLOCKS = ((RTOT / 16) * (DD / 64) * 32) / TB;  // 64
    const int GEMM16_BLOCKS  = ((RTOT / 16) * (SS / 16) * 32) / TB;  // 16

    // 1) RMSNorm
    rmsnorm_kernel<<<(RTOT * 32) / TB, TB, 0, stream>>>(x, norm_w, xn);

    // 2) Input projections + residual
    wmma_gemm_kernel<DD, DD2, 0><<<GEMM512_BLOCKS, TB, 0, stream>>>(xn, w_inp, b_inp, xproj);
    wmma_gemm_kernel<DD, DD2, 0><<<GEMM512_BLOCKS, TB, 0, stream>>>(mouth, w_inp, b_inp, mouthp);
    wmma_gemm_kernel<DD, DD2, 1><<<GEMM512_BLOCKS, TB, 0, stream>>>(xn, w_d, b_d, xres);

    // 3) Branches: (flipL, flipD) = (0,1), (1,0), (1,1), (0,0)
    const int flips[4][2] = {{0, 1}, {1, 0}, {1, 1}, {0, 0}};
    for (int br = 0; br < 4; ++br) {
        int fL = flips[br][0], fD = flips[br][1];
        flip_copy_kernel<<<ELEM_BLOCKS, TB, 0, stream>>>(mouthp, vb, fL, fD);
        conv_silu_kernel<<<ELEM_BLOCKS, TB, 0, stream>>>(xproj, conv_w, conv_b, hc, fL, fD);

        wmma_gemm_kernel<DD2, DD2, 0><<<GEMM512_BLOCKS, TB, 0, stream>>>(hc, fc1_w, fc1_b, g1x);
        wmma_gemm_kernel<DD2, DD2, 0><<<GEMM512_BLOCKS, TB, 0, stream>>>(vb, fc1_w, fc1_b, g1v);
        wmma_gemm_kernel<DD2, SS, 0><<<GEMM16_BLOCKS, TB, 0, stream>>>(hc, fc2_w, fc2_b, bmrows);
        wmma_gemm_kernel<DD2, SS, 0><<<GEMM16_BLOCKS, TB, 0, stream>>>(vb, fc2_w, fc2_b, erows);
        wmma_gemm_kernel<DD2, SS, 0><<<GEMM16_BLOCKS, TB, 0, stream>>>(hc, fc3_w, fc3_b, cmrows);

        dots_kernel<<<RTOT / TB, TB, 0, stream>>>(cmrows, bmrows, erows, cbv, cev);
        combine_kernel<<<ELEM_BLOCKS, TB, 0, stream>>>(g1x, g1v, hc, vb, cbv, cev,
                                                       combined, br == 0 ? 1 : 0);
    }

    // 4) combined *= 0.25 * x_residual; final projection to D
    scale_kernel<<<ELEM_BLOCKS, TB, 0, stream>>>(combined, xres);
    wmma_gemm_kernel<DD2, DD, 0><<<GEMM256_BLOCKS, TB, 0, stream>>>(combined, w_out, b_out, out);
}